// SimplifiedStatePath_24730421691117
// MI455X (gfx1250) — compile-verified
//
#include <hip/hip_runtime.h>
#include <hip/hip_bf16.h>

typedef __attribute__((ext_vector_type(2))) float v2f;
typedef __attribute__((ext_vector_type(8))) float v8f;

#define B_   4
#define T_   4096
#define DM   1024
#define DS   256
#define LCH  128            // scan chunk length
#define NCH  (T_ / LCH)     // 32 chunks

// ---------------------------------------------------------------------------
// GEMM1: gu[m, s] = gamma[s] * ( sum_k x[m,k] * W_in[k,s] + b_in[s] )
// M = B*T = 16384, N = DS = 256, K = DM = 1024.
// One wave computes a 64(M) x 16(N) tile via v_wmma_f32_16x16x4_f32.
// ---------------------------------------------------------------------------
__global__ __launch_bounds__(256) void gemm_in_kernel(
    const float* __restrict__ x, const float* __restrict__ W_in,
    const float* __restrict__ b_in, const float* __restrict__ gamma,
    float* __restrict__ gu)
{
  const int lane  = threadIdx.x & 31;
  const int wave  = threadIdx.x >> 5;
  const int gw    = blockIdx.x * 8 + wave;
  const int tilesN = DS / 16;                 // 16
  const int waveM = gw / tilesN;              // 0..255
  const int waveN = gw % tilesN;              // 0..15
  const int m0 = waveM * 64;
  const int n0 = waveN * 16;
  const int lm = lane & 15;                   // row (A) / col (B) within tile
  const int kk = (lane >> 4) << 1;            // K pair base: 0 or 2

  v8f acc0 = {}, acc1 = {}, acc2 = {}, acc3 = {};

  for (int k = 0; k < DM; k += 4) {
    const int kr = k + kk;
    v2f bfrag;
    bfrag.x = W_in[(size_t)(kr + 0) * DS + n0 + lm];
    bfrag.y = W_in[(size_t)(kr + 1) * DS + n0 + lm];
    const float* xrow = x + (size_t)(m0 + lm) * DM + kr;
    v2f a0 = *(const v2f*)(xrow + (size_t)0 * 16 * DM);
    v2f a1 = *(const v2f*)(xrow + (size_t)1 * 16 * DM);
    v2f a2 = *(const v2f*)(xrow + (size_t)2 * 16 * DM);
    v2f a3 = *(const v2f*)(xrow + (size_t)3 * 16 * DM);
    acc0 = __builtin_amdgcn_wmma_f32_16x16x4_f32(false, a0, false, bfrag, (short)0, acc0, false, false);
    acc1 = __builtin_amdgcn_wmma_f32_16x16x4_f32(false, a1, false, bfrag, (short)0, acc1, false, false);
    acc2 = __builtin_amdgcn_wmma_f32_16x16x4_f32(false, a2, false, bfrag, (short)0, acc2, false, false);
    acc3 = __builtin_amdgcn_wmma_f32_16x16x4_f32(false, a3, false, bfrag, (short)0, acc3, false, false);
  }

  // C/D layout: VGPR r, lanes 0-15 -> M=r, lanes 16-31 -> M=r+8; N = lane%16
  const int rbase = (lane >> 4) * 8;
  const int col   = n0 + lm;
  const float bi  = b_in[col];
  const float g   = gamma[col];
#pragma unroll
  for (int r = 0; r < 8; ++r) {
    gu[(size_t)(m0 +  0 + rbase + r) * DS + col] = g * (acc0[r] + bi);
    gu[(size_t)(m0 + 16 + rbase + r) * DS + col] = g * (acc1[r] + bi);
    gu[(size_t)(m0 + 32 + rbase + r) * DS + col] = g * (acc2[r] + bi);
    gu[(size_t)(m0 + 48 + rbase + r) * DS + col] = g * (acc3[r] + bi);
  }
}

// ---------------------------------------------------------------------------
// Scan phase A: per-chunk local end value (zero-init recurrence), read-only.
// chunkend[b,c,s] = sum_{j<L} a^(L-1-j) * gu[b, c*L+j, s]
// Block = (b, chunk); thread = state channel s.
// ---------------------------------------------------------------------------
__global__ __launch_bounds__(DS) void chunk_end_kernel(
    const float* __restrict__ lambda_p, const float* __restrict__ gu,
    float* __restrict__ chunkend)
{
  const int b = blockIdx.x / NCH;
  const int c = blockIdx.x % NCH;
  const int s = threadIdx.x;
  const float a = 1.0f / (1.0f + expf(-lambda_p[s]));
  size_t idx = ((size_t)(b * T_ + c * LCH)) * DS + s;
  float e = 0.0f;
#pragma unroll 4
  for (int j = 0; j < LCH; ++j) {
    e = fmaf(a, e, gu[idx]);
    idx += DS;
  }
  chunkend[(size_t)(b * NCH + c) * DS + s] = e;
}

// ---------------------------------------------------------------------------
// Scan phase B: per-channel carry scan across the 32 chunks.
// carry[b,c,s] = exact h entering chunk c (includes initial `state`).
// carry_{c+1} = a^L * carry_c + chunkend_c ; a^L = a^128 via 7 squarings.
// ---------------------------------------------------------------------------
__global__ __launch_bounds__(DS) void scan_carry_kernel(
    const float* __restrict__ lambda_p, const float* __restrict__ state,
    const float* __restrict__ chunkend, float* __restrict__ carrybuf)
{
  const int b = blockIdx.x;
  const int s = threadIdx.x;
  const float a = 1.0f / (1.0f + expf(-lambda_p[s]));

  float aL = a;
#pragma unroll
  for (int i = 0; i < 7; ++i) aL *= aL;   // a^(2^7) = a^LCH

  float carry = state[(size_t)b * DS + s];
  for (int c = 0; c < NCH; ++c) {
    carrybuf[(size_t)(b * NCH + c) * DS + s] = carry;
    carry = fmaf(aL, carry, chunkend[(size_t)(b * NCH + c) * DS + s]);
  }
}

// ---------------------------------------------------------------------------
// Scan phase C: re-run the exact recurrence per chunk seeded with its carry,
// overwriting gu with h.  Also emits h[:, T-1, :].
// ---------------------------------------------------------------------------
__global__ __launch_bounds__(DS) void rescan_kernel(
    const float* __restrict__ lambda_p, float* __restrict__ gu,
    const float* __restrict__ carrybuf, float* __restrict__ h_last)
{
  const int b = blockIdx.x / NCH;
  const int c = blockIdx.x % NCH;
  const int s = threadIdx.x;
  const float a = 1.0f / (1.0f + expf(-lambda_p[s]));
  size_t idx = ((size_t)(b * T_ + c * LCH)) * DS + s;
  float h = carrybuf[(size_t)(b * NCH + c) * DS + s];
#pragma unroll 4
  for (int j = 0; j < LCH; ++j) {
    h = fmaf(a, h, gu[idx]);
    gu[idx] = h;
    idx += DS;
  }
  if (c == NCH - 1) h_last[(size_t)b * DS + s] = h;
}

// ---------------------------------------------------------------------------
// GEMM2: y[m, d] = sum_s h[m,s] * W_out[s,d] + b_out[d]
// M = 16384, N = DM = 1024, K = DS = 256.  Wave computes 64x16 tile.
// ---------------------------------------------------------------------------
__global__ __launch_bounds__(256) void gemm_out_kernel(
    const float* __restrict__ h, const float* __restrict__ W_out,
    const float* __restrict__ b_out, float* __restrict__ y)
{
  const int lane  = threadIdx.x & 31;
  const int wave  = threadIdx.x >> 5;
  const int gw    = blockIdx.x * 8 + wave;
  const int tilesN = DM / 16;                 // 64
  const int waveM = gw / tilesN;              // 0..255
  const int waveN = gw % tilesN;              // 0..63
  const int m0 = waveM * 64;
  const int n0 = waveN * 16;
  const int lm = lane & 15;
  const int kk = (lane >> 4) << 1;

  v8f acc0 = {}, acc1 = {}, acc2 = {}, acc3 = {};

  for (int k = 0; k < DS; k += 4) {
    const int kr = k + kk;
    v2f bfrag;
    bfrag.x = W_out[(size_t)(kr + 0) * DM + n0 + lm];
    bfrag.y = W_out[(size_t)(kr + 1) * DM + n0 + lm];
    const float* hrow = h + (size_t)(m0 + lm) * DS + kr;
    v2f a0 = *(const v2f*)(hrow + (size_t)0 * 16 * DS);
    v2f a1 = *(const v2f*)(hrow + (size_t)1 * 16 * DS);
    v2f a2 = *(const v2f*)(hrow + (size_t)2 * 16 * DS);
    v2f a3 = *(const v2f*)(hrow + (size_t)3 * 16 * DS);
    acc0 = __builtin_amdgcn_wmma_f32_16x16x4_f32(false, a0, false, bfrag, (short)0, acc0, false, false);
    acc1 = __builtin_amdgcn_wmma_f32_16x16x4_f32(false, a1, false, bfrag, (short)0, acc1, false, false);
    acc2 = __builtin_amdgcn_wmma_f32_16x16x4_f32(false, a2, false, bfrag, (short)0, acc2, false, false);
    acc3 = __builtin_amdgcn_wmma_f32_16x16x4_f32(false, a3, false, bfrag, (short)0, acc3, false, false);
  }

  const int rbase = (lane >> 4) * 8;
  const int col   = n0 + lm;
  const float bo  = b_out[col];
#pragma unroll
  for (int r = 0; r < 8; ++r) {
    y[(size_t)(m0 +  0 + rbase + r) * DM + col] = acc0[r] + bo;
    y[(size_t)(m0 + 16 + rbase + r) * DM + col] = acc1[r] + bo;
    y[(size_t)(m0 + 32 + rbase + r) * DM + col] = acc2[r] + bo;
    y[(size_t)(m0 + 48 + rbase + r) * DM + col] = acc3[r] + bo;
  }
}

// ---------------------------------------------------------------------------
extern "C" void kernel_launch(void* const* d_in, const int* in_sizes, int n_in,
                              void* d_out, int out_size, void* d_ws, size_t ws_size,
                              hipStream_t stream) {
  const float* x        = (const float*)d_in[0];  // [B,T,DM]
  const float* state    = (const float*)d_in[1];  // [B,DS]
  const float* W_in     = (const float*)d_in[2];  // [DM,DS]
  const float* b_in     = (const float*)d_in[3];  // [DS]
  const float* lambda_p = (const float*)d_in[4];  // [DS]
  const float* gamma_p  = (const float*)d_in[5];  // [DS]
  const float* W_out    = (const float*)d_in[6];  // [DS,DM]
  const float* b_out    = (const float*)d_in[7];  // [DM]

  float* y      = (float*)d_out;                        // [B,T,DM]
  float* h_last = y + (size_t)B_ * T_ * DM;             // [B,DS]

  float* gu       = (float*)d_ws;                       // [B,T,DS] (becomes h)
  float* chunkend = gu + (size_t)B_ * T_ * DS;          // [B,NCH,DS]
  float* carry    = chunkend + (size_t)B_ * NCH * DS;   // [B,NCH,DS]

  // GEMM1: 4096 waves -> 512 blocks of 8 waves
  gemm_in_kernel<<<512, 256, 0, stream>>>(x, W_in, b_in, gamma_p, gu);

  // Chunked linear scan: chunk-ends -> carries -> exact seeded re-scan
  chunk_end_kernel<<<B_ * NCH, DS, 0, stream>>>(lambda_p, gu, chunkend);
  scan_carry_kernel<<<B_, DS, 0, stream>>>(lambda_p, state, chunkend, carry);
  rescan_kernel<<<B_ * NCH, DS, 0, stream>>>(lambda_p, gu, carry, h_last);

  // GEMM2: 16384 waves -> 2048 blocks of 8 waves
  gemm_out_kernel<<<2048, 256, 0, stream>>>(gu, W_out, b_out, y);
}